// BinOverlapPredictionFromMaxProjManualModule_85023172592447
// MI455X (gfx1250) — compile-verified
//
#include <hip/hip_runtime.h>
#include <hip/hip_bf16.h>
#include <float.h>

// ---------------------------------------------------------------------------
// Problem constants (from the reference): ptm (32,8,30,1,72,72) f32,
// mem_mask (32,8,30) bool. out = masked mean over mem of max over 72*72 feats.
// ---------------------------------------------------------------------------
#define NF    32
#define NS    8
#define NROW  (NF * NS)      // 256 output rows
#define NMEM  30
#define FEAT  (72 * 72)      // 5184 floats per (row, mem)
#define FEAT4 (FEAT / 4)     // 1296 float4 per (row, mem)

typedef __attribute__((ext_vector_type(2))) float v2f;
typedef __attribute__((ext_vector_type(8))) float v8f;

// ---------------------------------------------------------------------------
// Phase 1: streaming max-reduction. One block per (row, mem) pair.
// 159 MB total read — pure bandwidth problem. b128 loads, register v_max,
// wave32 shuffle reduce, cross-wave LDS reduce. RT (default) temporal hints:
// the 159 MB input fits the 192 MB L2, so graph-replay passes hit L2.
// ---------------------------------------------------------------------------
__global__ __launch_bounds__(256) void maxproj_kernel(const float* __restrict__ ptm,
                                                      float* __restrict__ pmax) {
    const int bid  = blockIdx.x;          // row * NMEM + mem, 0..7679
    const int tid  = threadIdx.x;
    const float4* __restrict__ src =
        reinterpret_cast<const float4*>(ptm) + (size_t)bid * FEAT4;

    float m = -FLT_MAX;
    // 1296 float4 across 256 threads: 5-6 iterations, plenty of loads in flight.
    #pragma unroll 2
    for (int i = tid; i < FEAT4; i += 256) {
        const float4 v = src[i];           // global_load_b128
        m = fmaxf(m, fmaxf(fmaxf(v.x, v.y), fmaxf(v.z, v.w)));
    }

    // wave32 butterfly reduction (5 steps)
    #pragma unroll
    for (int off = 16; off > 0; off >>= 1)
        m = fmaxf(m, __shfl_xor(m, off, 32));

    __shared__ float smax[8];              // 8 waves per 256-thread block
    const int wave = tid >> 5;
    const int lane = tid & 31;
    if (lane == 0) smax[wave] = m;
    __syncthreads();
    if (tid == 0) {
        float r = smax[0];
        #pragma unroll
        for (int w = 1; w < 8; ++w) r = fmaxf(r, smax[w]);
        pmax[bid] = r;
    }
}

// ---------------------------------------------------------------------------
// Phase 2: masked mean over mems as a 256x30 masked GEMV on the WMMA path.
//   sum_val  = (P .* M) x ones   via V_WMMA_F32_16X16X4_F32 accumulation
//   sum_mask =  M        x ones  via a second WMMA chain
// One wave (32 threads) handles 16 rows; K = 30 padded to 32 (zeros).
// A-matrix f32 16x4 layout: lanes 0-15 hold K=k0,k0+1; lanes 16-31 hold
// K=k0+2,k0+3 (ISA 7.12.2). D column N=0 lives in lane 0 (M=0..7, VGPR 0..7)
// and lane 16 (M=8..15).
// ---------------------------------------------------------------------------
__global__ __launch_bounds__(32) void masked_mean_wmma(const float* __restrict__ pmax,
                                                       const unsigned char* __restrict__ mask,
                                                       float* __restrict__ out) {
    const int lane  = threadIdx.x;             // 0..31, EXEC all-ones for WMMA
    const int row0  = blockIdx.x * 16;         // 16 rows per wave
    const int mrow  = row0 + (lane & 15);      // this lane's A-matrix row
    const int kbase = (lane < 16) ? 0 : 2;     // K pair carried by this lane

    v8f acc_v = {};                            // masked-value row sums
    v8f acc_m = {};                            // mask-count row sums
    v2f bones; bones.x = 1.0f; bones.y = 1.0f; // B = ones(4x16), any layout

    #pragma unroll
    for (int k0 = 0; k0 < 32; k0 += 4) {
        v2f av; av.x = 0.0f; av.y = 0.0f;      // zero-pad K = 30,31
        v2f am; am.x = 0.0f; am.y = 0.0f;
        const int k = k0 + kbase;
        if (k < NMEM) {
            const float m0 = mask[mrow * NMEM + k] ? 1.0f : 0.0f;
            am.x = m0;
            av.x = pmax[mrow * NMEM + k] * m0;
        }
        if (k + 1 < NMEM) {
            const float m1 = mask[mrow * NMEM + k + 1] ? 1.0f : 0.0f;
            am.y = m1;
            av.y = pmax[mrow * NMEM + k + 1] * m1;
        }
        // D = A x ones + C : every D column holds the K-partial row sums.
        acc_v = __builtin_amdgcn_wmma_f32_16x16x4_f32(
            /*neg_a=*/false, av, /*neg_b=*/false, bones,
            /*c_mod=*/(short)0, acc_v, /*reuse_a=*/false, /*reuse_b=*/false);
        acc_m = __builtin_amdgcn_wmma_f32_16x16x4_f32(
            false, am, false, bones, (short)0, acc_m, false, false);
    }

    // Column N=0 of D: lane 0 -> rows row0+0..7 (VGPR 0..7),
    //                  lane 16 -> rows row0+8..15.
    if (lane == 0) {
        #pragma unroll
        for (int r = 0; r < 8; ++r) out[row0 + r] = acc_v[r] / acc_m[r];
    } else if (lane == 16) {
        #pragma unroll
        for (int r = 0; r < 8; ++r) out[row0 + 8 + r] = acc_v[r] / acc_m[r];
    }
}

// ---------------------------------------------------------------------------
// d_in[0]: ptm   (float32, 32*8*30*1*72*72 elems)
// d_in[1]: mem_mask (bool -> 1 byte/elem, 32*8*30 elems)
// d_out  : 256 float32
// d_ws   : scratch; uses 7680 * 4 B for the per-(row,mem) maxima
// ---------------------------------------------------------------------------
extern "C" void kernel_launch(void* const* d_in, const int* in_sizes, int n_in,
                              void* d_out, int out_size, void* d_ws, size_t ws_size,
                              hipStream_t stream) {
    const float* ptm          = (const float*)d_in[0];
    const unsigned char* mask = (const unsigned char*)d_in[1]; // jax bool_ = 1 byte
    float* out                = (float*)d_out;
    float* pmax               = (float*)d_ws;                  // 7680 floats

    maxproj_kernel<<<dim3(NROW * NMEM), dim3(256), 0, stream>>>(ptm, pmax);
    masked_mean_wmma<<<dim3(NROW / 16), dim3(32), 0, stream>>>(pmax, mask, out);
}